// better_loss_29231547416732
// MI455X (gfx1250) — compile-verified
//
#include <hip/hip_runtime.h>
#include <math.h>

typedef __attribute__((ext_vector_type(2))) float v2f;
typedef __attribute__((ext_vector_type(8))) float v8f;

#define B_ 64
#define T_ 10
#define K_ 3
#define F_ 3
#define TFXY 122880.0f
#define LDR 66  // LDS row stride in words for 64-wide matrices (even -> b64 aligned, bank friendly)

__device__ __forceinline__ float elu1(float z) { return z > 0.f ? z + 1.f : expf(z); }

// ---------------------------------------------------------------------------
// Prep: process triangular factors (tril, diag -> elu+1), build
//   gAx = P_x^T (A-matrix for x mode product)
//   gPy = P_y   (== A_y^T, the B-matrix for the y mode product)
//   gAt = P_t^T, gAf = P_f^T  (small factors for pointwise t/f mixing)
//   gLogdet[k]
// ---------------------------------------------------------------------------
__global__ __launch_bounds__(256) void prep_kernel(
    const float* __restrict__ L_t, const float* __restrict__ L_f,
    const float* __restrict__ L_x, const float* __restrict__ L_y,
    float* __restrict__ gAx, float* __restrict__ gPy,
    float* __restrict__ gAt, float* __restrict__ gAf,
    float* __restrict__ gLogdet)
{
  const int tid = threadIdx.x;
  for (int k = 0; k < K_; ++k) {
    for (int idx = tid; idx < 64 * 64; idx += 256) {
      int i = idx >> 6, j = idx & 63;
      float lx = L_x[k * 4096 + idx];
      float px = (j < i) ? lx : (j == i ? elu1(lx) : 0.f);
      gAx[k * 4096 + j * 64 + i] = px;           // transpose: A_x = P_x^T
      float ly = L_y[k * 4096 + idx];
      float py = (j < i) ? ly : (j == i ? elu1(ly) : 0.f);
      gPy[k * 4096 + idx] = py;                  // keep P_y (B matrix = A_y^T = P_y)
    }
    for (int idx = tid; idx < 100; idx += 256) {
      int i = idx / 10, j = idx % 10;
      float lt = L_t[k * 100 + idx];
      float pt = (j < i) ? lt : (j == i ? elu1(lt) : 0.f);
      gAt[k * 100 + j * 10 + i] = pt;            // A_t = P_t^T
    }
    for (int idx = tid; idx < 9; idx += 256) {
      int i = idx / 3, j = idx % 3;
      float lf = L_f[k * 9 + idx];
      float pf = (j < i) ? lf : (j == i ? elu1(lf) : 0.f);
      gAf[k * 9 + j * 3 + i] = pf;               // A_f = P_f^T
    }
    if (tid == 0) {
      float st = 0.f, sf = 0.f, sx = 0.f, sy = 0.f;
      for (int i = 0; i < 10; ++i) st += logf(elu1(L_t[k * 100 + i * 11]));
      for (int i = 0; i < 3;  ++i) sf += logf(elu1(L_f[k * 9 + i * 4]));
      for (int i = 0; i < 64; ++i) sx += logf(elu1(L_x[k * 4096 + i * 65]));
      for (int i = 0; i < 64; ++i) sy += logf(elu1(L_y[k * 4096 + i * 65]));
      gLogdet[k] = TFXY * (st / 10.f + sf / 3.f + sx / 64.f + sy / 64.f);
    }
  }
}

// ---------------------------------------------------------------------------
// Main: one block per (b,k). For each of the 30 (t,f) output slabs:
//   V = sum_{t',f'} At[t,t']Af[f,f'] (pred - target)   (pointwise in x,y)
//   T1 = A_x @ V      (64x64x64 fp32 GEMM via v_wmma_f32_16x16x4_f32)
//   Lv = T1 @ P_y     (same)
//   mahal += ||Lv||_F^2  accumulated straight from WMMA D registers.
// ---------------------------------------------------------------------------
__global__ __launch_bounds__(256) void mahal_kernel(
    const float* __restrict__ pred, const float* __restrict__ target,
    const float* __restrict__ gAx, const float* __restrict__ gPy,
    const float* __restrict__ gAt, const float* __restrict__ gAf,
    float* __restrict__ gMahal)
{
  __shared__ float sAx[64 * LDR];
  __shared__ float sPy[64 * LDR];
  __shared__ float sV [64 * LDR];
  __shared__ float sT1[64 * LDR];
  __shared__ float sW [30 * 30];
  __shared__ float sRed[256];

  const int tid    = threadIdx.x;
  const int lane   = tid & 31;
  const int wave   = tid >> 5;
  const int lane15 = lane & 15;
  const int hi     = lane >> 4;   // 0: lanes 0-15 (K0/K1), 1: lanes 16-31 (K2/K3)

  const int b = blockIdx.x / K_;
  const int k = blockIdx.x % K_;

  for (int idx = tid; idx < 64 * 64; idx += 256) {
    int r = idx >> 6, c = idx & 63;
    sAx[r * LDR + c] = gAx[k * 4096 + idx];
    sPy[r * LDR + c] = gPy[k * 4096 + idx];
  }
  for (int idx = tid; idx < 900; idx += 256) {
    int i = idx / 30, j = idx - i * 30;
    int t = i / 3, f = i - t * 3;
    int t2 = j / 3, f2 = j - t2 * 3;
    sW[idx] = gAt[k * 100 + t * 10 + t2] * gAf[k * 9 + f * 3 + f2];
  }
  __syncthreads();

  float mah = 0.f;

  for (int slab = 0; slab < 30; ++slab) {
    // ---- V build: weighted residual mix over (t',f'), pointwise in (x,y)
    float v[16];
#pragma unroll
    for (int e = 0; e < 16; ++e) v[e] = 0.f;
    for (int j = 0; j < 30; ++j) {
      float w = sW[slab * 30 + j];
      if (w != 0.f) {  // upper-triangular sparsity of A_t, A_f (uniform branch)
        int t2 = j / 3, f2 = j - t2 * 3;
        const float* pp = pred   + (size_t)((((b * T_ + t2) * K_ + k) * F_ + f2)) * 4096;
        const float* tg = target + (size_t)(((b * T_ + t2) * F_ + f2)) * 4096;
#pragma unroll
        for (int e = 0; e < 16; ++e) {
          int idx = tid + 256 * e;
          v[e] = fmaf(w, pp[idx] - tg[idx], v[e]);
        }
      }
    }
#pragma unroll
    for (int e = 0; e < 16; ++e) {
      int idx = tid + 256 * e;
      sV[(idx >> 6) * LDR + (idx & 63)] = v[e];
    }
    __syncthreads();

    // ---- GEMM1: T1 = Ax @ V   (8 waves x 2 tiles = 16 16x16 tiles)
    for (int tile = wave * 2; tile < wave * 2 + 2; ++tile) {
      int ti = tile >> 2, tj = tile & 3;
      v8f acc = {};
#pragma unroll
      for (int k0 = 0; k0 < 64; k0 += 4) {
        int kk = k0 + 2 * hi;
        v2f a = *(const v2f*)&sAx[(ti * 16 + lane15) * LDR + kk];  // A[row][kk], A[row][kk+1]
        v2f bb;
        bb.x = sV[(kk    ) * LDR + tj * 16 + lane15];              // B[kk][col]
        bb.y = sV[(kk + 1) * LDR + tj * 16 + lane15];              // B[kk+1][col]
        acc = __builtin_amdgcn_wmma_f32_16x16x4_f32(false, a, false, bb,
                                                    (short)0, acc, false, false);
      }
#pragma unroll
      for (int r = 0; r < 8; ++r)
        sT1[(ti * 16 + r + 8 * hi) * LDR + tj * 16 + lane15] = acc[r];
    }
    __syncthreads();

    // ---- GEMM2: Lv = T1 @ Py ; accumulate squared Frobenius norm
    for (int tile = wave * 2; tile < wave * 2 + 2; ++tile) {
      int ti = tile >> 2, tj = tile & 3;
      v8f acc = {};
#pragma unroll
      for (int k0 = 0; k0 < 64; k0 += 4) {
        int kk = k0 + 2 * hi;
        v2f a = *(const v2f*)&sT1[(ti * 16 + lane15) * LDR + kk];
        v2f bb;
        bb.x = sPy[(kk    ) * LDR + tj * 16 + lane15];
        bb.y = sPy[(kk + 1) * LDR + tj * 16 + lane15];
        acc = __builtin_amdgcn_wmma_f32_16x16x4_f32(false, a, false, bb,
                                                    (short)0, acc, false, false);
      }
#pragma unroll
      for (int r = 0; r < 8; ++r) mah = fmaf(acc[r], acc[r], mah);
    }
    __syncthreads();  // protect sV/sT1 before next slab overwrites them
  }

  sRed[tid] = mah;
  __syncthreads();
  for (int s = 128; s > 0; s >>= 1) {
    if (tid < s) sRed[tid] += sRed[tid + s];
    __syncthreads();
  }
  if (tid == 0) gMahal[b * K_ + k] = sRed[0];
}

// ---------------------------------------------------------------------------
// Finalize: nll = -logsumexp_k( C0 - 0.5*mahal + logdet + logw ), mean over b.
// ---------------------------------------------------------------------------
__global__ __launch_bounds__(64) void finalize_kernel(
    const float* __restrict__ gMahal, const float* __restrict__ gLogdet,
    const float* __restrict__ logw, float* __restrict__ out)
{
  __shared__ float red[64];
  const int b = threadIdx.x;
  const float C0 = (float)(-0.5 * 122880.0 * 1.8378770664093453);  // -tfxy/2 * ln(2*pi)
  float n[K_];
  float m = -INFINITY;
  for (int k = 0; k < K_; ++k) {
    n[k] = C0 - 0.5f * gMahal[b * K_ + k] + gLogdet[k] + logw[b * K_ + k];
    m = fmaxf(m, n[k]);
  }
  float s = 0.f;
  for (int k = 0; k < K_; ++k) s += expf(n[k] - m);
  red[b] = -(m + logf(s));
  __syncthreads();
  for (int t = 32; t > 0; t >>= 1) {
    if (b < t) red[b] += red[b + t];
    __syncthreads();
  }
  if (b == 0) out[0] = red[0] * (1.f / 64.f);
}

extern "C" void kernel_launch(void* const* d_in, const int* in_sizes, int n_in,
                              void* d_out, int out_size, void* d_ws, size_t ws_size,
                              hipStream_t stream) {
  (void)in_sizes; (void)n_in; (void)out_size; (void)ws_size;
  const float* pred   = (const float*)d_in[0];
  const float* target = (const float*)d_in[1];
  const float* logw   = (const float*)d_in[2];
  const float* L_t    = (const float*)d_in[3];
  const float* L_f    = (const float*)d_in[4];
  const float* L_x    = (const float*)d_in[5];
  const float* L_y    = (const float*)d_in[6];

  float* ws  = (float*)d_ws;
  float* gAx = ws;               // K*64*64 = 12288
  float* gPy = gAx + 12288;      // 12288
  float* gAt = gPy + 12288;      // 300
  float* gAf = gAt + 300;        // 27
  float* gLd = gAf + 27;         // 3
  float* gMh = gLd + 3;          // 192

  prep_kernel<<<1, 256, 0, stream>>>(L_t, L_f, L_x, L_y, gAx, gPy, gAt, gAf, gLd);
  mahal_kernel<<<B_ * K_, 256, 0, stream>>>(pred, target, gAx, gPy, gAt, gAf, gMh);
  finalize_kernel<<<1, 64, 0, stream>>>(gMh, gLd, logw, (float*)d_out);
}